// GAT_3_64072322122027
// MI455X (gfx1250) — compile-verified
//
#include <hip/hip_runtime.h>
#include <math.h>

#define NN 50000
#define NE 800000
#define NG 64
#define F_IN 128
#define HIDD 64
#define NHEAD 4
#define FEAT (NHEAD * HIDD) /* 256 */
#define OUTC 10
#define NEG_SLOPE 0.2f

typedef __attribute__((ext_vector_type(2))) float v2f;
typedef __attribute__((ext_vector_type(8))) float v8f;

// ---------------------------------------------------------------------------
// WMMA fp32 GEMM: C[M,256] = A[M,K] @ B[K,256], K compile-time (128 or 256).
// One 16(M) x 64(N) strip per wave: 4 independent v8f accumulators so the
// 4 WMMAs per K-step have no RAW chain; A fragment reused 4x.
// All strides are compile-time constants -> constant-offset loads, no
// per-iteration 64-bit multiplies.  Loads for a K-step are issued as one
// batch before the 4 WMMAs so the scheduler can clause them and overlap
// with the matrix pipe.
// A-frag (16x4 f32): lanes 0-15 -> M=lane, VGPR0=K0,VGPR1=K1;
//                    lanes 16-31 -> M=lane-16, VGPR0=K2,VGPR1=K3.
// B-frag (4x16 f32): VGPR0 = K0 (lanes0-15)/K2 (lanes16-31), VGPR1 = K1/K3.
// C/D: VGPR r -> (M = r + 8*(lane>=16), N = lane&15).
// ---------------------------------------------------------------------------
template <int K>
__global__ __launch_bounds__(128) void gemm_wmma_f32_kernel(
    const float* __restrict__ A, const float* __restrict__ B,
    float* __restrict__ C) {
  constexpr int NC = 256;
  const int lane = threadIdx.x & 31;
  const int wave = threadIdx.x >> 5;
  const int tile = blockIdx.x * 4 + wave;  // grid sized exactly (EXEC all-ones)
  const int mt = tile >> 2;                // 4 N-strips of 64
  const int ns = tile & 3;
  const int half = lane >> 4;  // 0: K pair 0/1, 1: K pair 2/3
  const int l15 = lane & 15;
  const int nbase = ns * 64;

  const float* __restrict__ Ap = A + (long)(mt * 16 + l15) * K + half * 2;
  const float* __restrict__ Bp = B + (half * 2) * NC + nbase + l15;

  v8f a0 = {0.f, 0.f, 0.f, 0.f, 0.f, 0.f, 0.f, 0.f};
  v8f a1 = a0, a2 = a0, a3 = a0;

#pragma unroll 4
  for (int k0 = 0; k0 < K; k0 += 4) {
    v2f a;
    a.x = Ap[0];
    a.y = Ap[1];
    // batch all 8 B loads before any WMMA (constant offsets off Bp)
    float b0x = Bp[0], b1x = Bp[16], b2x = Bp[32], b3x = Bp[48];
    float b0y = Bp[NC], b1y = Bp[NC + 16], b2y = Bp[NC + 32], b3y = Bp[NC + 48];
    v2f b0, b1, b2, b3;
    b0.x = b0x; b0.y = b0y;
    b1.x = b1x; b1.y = b1y;
    b2.x = b2x; b2.y = b2y;
    b3.x = b3x; b3.y = b3y;
    a0 = __builtin_amdgcn_wmma_f32_16x16x4_f32(false, a, false, b0, (short)0, a0, false, false);
    a1 = __builtin_amdgcn_wmma_f32_16x16x4_f32(false, a, false, b1, (short)0, a1, false, false);
    a2 = __builtin_amdgcn_wmma_f32_16x16x4_f32(false, a, false, b2, (short)0, a2, false, false);
    a3 = __builtin_amdgcn_wmma_f32_16x16x4_f32(false, a, false, b3, (short)0, a3, false, false);
    Ap += 4;
    Bp += 4 * NC;
  }

  float* __restrict__ Crow = C + (long)(mt * 16 + half * 8) * NC + nbase + l15;
#pragma unroll
  for (int r = 0; r < 8; ++r) {
    float* p = Crow + (long)r * NC;
    p[0] = a0[r];
    p[16] = a1[r];
    p[32] = a2[r];
    p[48] = a3[r];
  }
}

// ---------------------------------------------------------------------------
// el/er attention projections: el[n,h] = sum_d feat[n,h,d]*attn_l[h,d]
// ---------------------------------------------------------------------------
__global__ void el_er_kernel(const float* __restrict__ feat,
                             const float* __restrict__ al,
                             const float* __restrict__ ar,
                             float* __restrict__ el, float* __restrict__ er) {
  int idx = blockIdx.x * blockDim.x + threadIdx.x;
  if (idx >= NN * NHEAD) return;
  int n = idx >> 2, h = idx & 3;
  const float* f = feat + (long)n * FEAT + h * HIDD;
  const float* L = al + h * HIDD;
  const float* R = ar + h * HIDD;
  float sl = 0.f, sr = 0.f;
#pragma unroll 8
  for (int d = 0; d < HIDD; ++d) {
    float v = f[d];
    sl += v * L[d];
    sr += v * R[d];
  }
  el[idx] = sl;
  er[idx] = sr;
}

__global__ void fill_kernel(float* __restrict__ p, float v, int n) {
  int i = blockIdx.x * blockDim.x + threadIdx.x;
  if (i < n) p[i] = v;
}

// ordered-int trick for float atomic max (init with -inf)
__device__ inline void atomicMaxFloat(float* addr, float val) {
  int iv = __float_as_int(val);
  if (iv >= 0)
    atomicMax((int*)addr, iv);
  else
    atomicMin((unsigned int*)addr, (unsigned int)iv);
}

// e = leaky_relu(el[src] + er[dst]); store e; segment max over dst
__global__ void edge_max_kernel(const float* __restrict__ el,
                                const float* __restrict__ er,
                                const int* __restrict__ src,
                                const int* __restrict__ dst,
                                float* __restrict__ ebuf,
                                float* __restrict__ mmax) {
  int idx = blockIdx.x * blockDim.x + threadIdx.x;
  if (idx >= NE * NHEAD) return;
  int e = idx >> 2, h = idx & 3;
  int s = src[e], t = dst[e];
  float v = el[s * NHEAD + h] + er[t * NHEAD + h];
  v = (v > 0.f) ? v : NEG_SLOPE * v;
  ebuf[idx] = v;
  atomicMaxFloat(&mmax[t * NHEAD + h], v);
}

// ex = exp(e - m[dst]); overwrite ebuf; segment sum over dst
__global__ void edge_exp_kernel(float* __restrict__ ebuf,
                                const float* __restrict__ mmax,
                                const int* __restrict__ dst,
                                float* __restrict__ ssum) {
  int idx = blockIdx.x * blockDim.x + threadIdx.x;
  if (idx >= NE * NHEAD) return;
  int e = idx >> 2, h = idx & 3;
  int t = dst[e];
  float ex = expf(ebuf[idx] - mmax[t * NHEAD + h]);
  ebuf[idx] = ex;
  atomicAdd(&ssum[t * NHEAD + h], ex);
}

// alpha = ex / ssum[dst]  (in place; hoists the divide out of the scatter)
__global__ void edge_alpha_kernel(float* __restrict__ ebuf,
                                  const float* __restrict__ ssum,
                                  const int* __restrict__ dst) {
  int idx = blockIdx.x * blockDim.x + threadIdx.x;
  if (idx >= NE * NHEAD) return;
  int e = idx >> 2, h = idx & 3;
  ebuf[idx] = ebuf[idx] / ssum[dst[e] * NHEAD + h];
}

// one block per edge; 256 lanes cover the H*D=256 message elements
__global__ __launch_bounds__(256) void scatter_kernel(
    const float* __restrict__ alpha, const int* __restrict__ src,
    const int* __restrict__ dst, const float* __restrict__ feat,
    float* __restrict__ agg) {
  int e = blockIdx.x;
  int hd = threadIdx.x;  // h*64 + d
  int h = hd >> 6;
  int s = src[e], t = dst[e];
  float a = alpha[e * NHEAD + h];
  atomicAdd(&agg[(long)t * FEAT + hd], a * feat[(long)s * FEAT + hd]);
}

// elu(agg + bias) in place
__global__ void elu_bias_kernel(float* __restrict__ agg,
                                const float* __restrict__ bias) {
  long idx = (long)blockIdx.x * blockDim.x + threadIdx.x;
  if (idx >= (long)NN * FEAT) return;
  int hd = (int)(idx & (FEAT - 1));
  float v = agg[idx] + bias[hd];
  agg[idx] = (v > 0.f) ? v : expm1f(v);
}

// mean over heads, then sum-pool per graph
__global__ void pool_kernel(const float* __restrict__ h2,
                            const int* __restrict__ gid,
                            float* __restrict__ pooled) {
  int idx = blockIdx.x * blockDim.x + threadIdx.x;
  if (idx >= NN * HIDD) return;
  int n = idx >> 6, d = idx & 63;
  const float* p = h2 + (long)n * FEAT + d;
  float m = 0.25f * (p[0] + p[HIDD] + p[2 * HIDD] + p[3 * HIDD]);
  atomicAdd(&pooled[gid[n] * HIDD + d], m);
}

// out[g,o] = pooled[g,:] @ fcW[:,o] + fcb[o]
__global__ void fc_kernel(const float* __restrict__ pooled,
                          const float* __restrict__ fcW,
                          const float* __restrict__ fcb,
                          float* __restrict__ out) {
  int idx = blockIdx.x * blockDim.x + threadIdx.x;
  if (idx >= NG * OUTC) return;
  int g = idx / OUTC, o = idx % OUTC;
  float acc = fcb[o];
#pragma unroll 8
  for (int d = 0; d < HIDD; ++d) acc += pooled[g * HIDD + d] * fcW[d * OUTC + o];
  out[idx] = acc;
}

extern "C" void kernel_launch(void* const* d_in, const int* in_sizes, int n_in,
                              void* d_out, int out_size, void* d_ws,
                              size_t ws_size, hipStream_t stream) {
  const float* x = (const float*)d_in[0];
  const int* src = (const int*)d_in[1];
  const int* dst = (const int*)d_in[2];
  const int* gid = (const int*)d_in[3];
  const float* W1 = (const float*)d_in[4];
  const float* al1 = (const float*)d_in[5];
  const float* ar1 = (const float*)d_in[6];
  const float* b1 = (const float*)d_in[7];
  const float* W2 = (const float*)d_in[8];
  const float* al2 = (const float*)d_in[9];
  const float* ar2 = (const float*)d_in[10];
  const float* b2 = (const float*)d_in[11];
  const float* fcW = (const float*)d_in[12];
  const float* fcb = (const float*)d_in[13];
  float* out = (float*)d_out;

  // workspace carve-up (floats)
  float* ws = (float*)d_ws;
  float* feat = ws;                      // NN*FEAT
  float* agg = feat + (long)NN * FEAT;   // NN*FEAT
  float* el = agg + (long)NN * FEAT;     // NN*NHEAD
  float* er = el + NN * NHEAD;           // NN*NHEAD
  float* mmax = er + NN * NHEAD;         // NN*NHEAD
  float* ssum = mmax + NN * NHEAD;       // NN*NHEAD
  float* ebuf = ssum + NN * NHEAD;       // NE*NHEAD
  float* pooled = ebuf + (long)NE * NHEAD;  // NG*HIDD

  const int gemm_blocks = (NN / 16) * (FEAT / 64) / 4;  // 3125, exact
  const int nh_blocks = (NN * NHEAD + 255) / 256;
  const int eh_blocks = (NE * NHEAD + 255) / 256;
  const int nf_blocks = (NN * FEAT + 255) / 256;
  const int nd_blocks = (NN * HIDD + 255) / 256;

  // ---------------- layer 1 ----------------
  gemm_wmma_f32_kernel<F_IN><<<gemm_blocks, 128, 0, stream>>>(x, W1, feat);
  el_er_kernel<<<nh_blocks, 256, 0, stream>>>(feat, al1, ar1, el, er);
  fill_kernel<<<nh_blocks, 256, 0, stream>>>(mmax, -INFINITY, NN * NHEAD);
  hipMemsetAsync(ssum, 0, (size_t)NN * NHEAD * sizeof(float), stream);
  hipMemsetAsync(agg, 0, (size_t)NN * FEAT * sizeof(float), stream);
  edge_max_kernel<<<eh_blocks, 256, 0, stream>>>(el, er, src, dst, ebuf, mmax);
  edge_exp_kernel<<<eh_blocks, 256, 0, stream>>>(ebuf, mmax, dst, ssum);
  edge_alpha_kernel<<<eh_blocks, 256, 0, stream>>>(ebuf, ssum, dst);
  scatter_kernel<<<NE, 256, 0, stream>>>(ebuf, src, dst, feat, agg);
  elu_bias_kernel<<<nf_blocks, 256, 0, stream>>>(agg, b1);  // agg = h1

  // ---------------- layer 2 ----------------
  gemm_wmma_f32_kernel<FEAT><<<gemm_blocks, 128, 0, stream>>>(agg, W2, feat);
  el_er_kernel<<<nh_blocks, 256, 0, stream>>>(feat, al2, ar2, el, er);
  fill_kernel<<<nh_blocks, 256, 0, stream>>>(mmax, -INFINITY, NN * NHEAD);
  hipMemsetAsync(ssum, 0, (size_t)NN * NHEAD * sizeof(float), stream);
  hipMemsetAsync(agg, 0, (size_t)NN * FEAT * sizeof(float), stream);  // h1 consumed
  edge_max_kernel<<<eh_blocks, 256, 0, stream>>>(el, er, src, dst, ebuf, mmax);
  edge_exp_kernel<<<eh_blocks, 256, 0, stream>>>(ebuf, mmax, dst, ssum);
  edge_alpha_kernel<<<eh_blocks, 256, 0, stream>>>(ebuf, ssum, dst);
  scatter_kernel<<<NE, 256, 0, stream>>>(ebuf, src, dst, feat, agg);
  elu_bias_kernel<<<nf_blocks, 256, 0, stream>>>(agg, b2);  // agg = h2 [N,4,64]

  // ---------------- readout ----------------
  hipMemsetAsync(pooled, 0, (size_t)NG * HIDD * sizeof(float), stream);
  pool_kernel<<<nd_blocks, 256, 0, stream>>>(agg, gid, pooled);
  fc_kernel<<<(NG * OUTC + 255) / 256, 256, 0, stream>>>(pooled, fcW, fcb, out);
}